// GnnKernel_38328288150249
// MI455X (gfx1250) — compile-verified
//
#include <hip/hip_runtime.h>

// GraphConv (PyG, aggr='add') on MI455X / gfx1250:
//   out = scatter_add(x[src] -> dst) @ W_rel^T + b_rel + x @ W_root^T
//
// Phase 1: zero agg (workspace)
// Phase 2: wave-per-edge gather + fp32 global atomic scatter-add (L2-resident)
// Phase 3: fused dual GEMM with V_WMMA_F32_16X16X4_F32 (fp32-exact),
//          A-operands staged in LDS, W matrices streamed from L2.

typedef __attribute__((ext_vector_type(2))) float v2f;
typedef __attribute__((ext_vector_type(4))) float v4f;
typedef __attribute__((ext_vector_type(8))) float v8f;

#define D 128   // D_IN == D_OUT == 128

// ---------------------------------------------------------------- zero agg
__global__ void gnn_zero_kernel(float* __restrict__ p, int n4) {
    int i = blockIdx.x * blockDim.x + threadIdx.x;
    if (i < n4) ((v4f*)p)[i] = (v4f){0.f, 0.f, 0.f, 0.f};
}

// ------------------------------------------------- edge gather/scatter-add
// One wave32 per edge; lane handles 4 contiguous features (float4 load,
// 4x global_atomic_add_f32). x rows and agg rows are L2-resident (51 MB each
// vs 192 MB L2), so this is an L2-bandwidth/atomic-throughput phase.
__global__ void __launch_bounds__(256)
gnn_scatter_kernel(const float* __restrict__ x,
                   const int* __restrict__ eidx,   // [2, E]: row0=src, row1=dst
                   float* __restrict__ agg, int E) {
    int t = blockIdx.x * blockDim.x + threadIdx.x;
    int e    = t >> 5;
    int lane = t & 31;
    if (e >= E) return;
    int src = eidx[e];
    int dst = eidx[E + e];
    v4f v = *(const v4f*)(x + (long)src * D + lane * 4);
    float* a = agg + (long)dst * D + lane * 4;
    atomicAdd(a + 0, v.x);
    atomicAdd(a + 1, v.y);
    atomicAdd(a + 2, v.z);
    atomicAdd(a + 3, v.w);
}

// ------------------------------------------------------- fused WMMA GEMM
// Block = 256 threads = 8 waves; block owns a 16-row output tile (16 x 128).
// Wave w owns output columns [16w, 16w+16).
// Per ISA 7.12.2 (32-bit A 16x4): lane l holds row M=l%16, K = {kh, kh+1}
// with kh = 2*(l/16). B (4x16, B[k][n] = W[n][k]): lane l holds col N=l%16,
// K = {kh, kh+1} -> contiguous float2 from W row n.
// C/D: VGPR r, lanes 0-15 -> M=r, lanes 16-31 -> M=r+8; N = l%16.
__global__ void __launch_bounds__(256)
gnn_gemm_kernel(const float* __restrict__ agg, const float* __restrict__ x,
                const float* __restrict__ Wrel, const float* __restrict__ brel,
                const float* __restrict__ Wroot, float* __restrict__ out) {
    __shared__ float sA[16 * D];   // agg rows for this tile
    __shared__ float sX[16 * D];   // x rows for this tile

    const int tid = threadIdx.x;
    const int m0  = blockIdx.x * 16;

    // Cooperative stage: 2 x 16x128 fp32 = 16 KB into LDS.
    const v4f* gA = (const v4f*)(agg + (long)m0 * D);
    const v4f* gX = (const v4f*)(x   + (long)m0 * D);
    for (int i = tid; i < 16 * D / 4; i += 256) {
        ((v4f*)sA)[i] = gA[i];
        ((v4f*)sX)[i] = gX[i];
    }
    __syncthreads();

    const int wave = tid >> 5;
    const int lane = tid & 31;
    const int n0   = wave * 16;       // output column tile for this wave
    const int row  = lane & 15;       // M for A-frag / N for B-frag
    const int kh   = (lane >> 4) * 2; // K sub-offset: 0 or 2

    const float* sap    = sA + row * D + kh;
    const float* sxp    = sX + row * D + kh;
    const float* wrelp  = Wrel  + (long)(n0 + row) * D + kh;
    const float* wrootp = Wroot + (long)(n0 + row) * D + kh;

    v8f acc = {};
#pragma unroll 4
    for (int k0 = 0; k0 < D; k0 += 4) {
        v2f a  = *(const v2f*)(sap + k0);       // ds_load_b64
        v2f b  = *(const v2f*)(wrelp + k0);     // global_load_b64 (L2-hot)
        acc = __builtin_amdgcn_wmma_f32_16x16x4_f32(
                  false, a, false, b, (short)0, acc, false, false);
        v2f a2 = *(const v2f*)(sxp + k0);
        v2f b2 = *(const v2f*)(wrootp + k0);
        acc = __builtin_amdgcn_wmma_f32_16x16x4_f32(
                  false, a2, false, b2, (short)0, acc, false, false);
    }

    const float bias = brel[n0 + row];
    const int mbase  = m0 + (lane >> 4) * 8;
#pragma unroll
    for (int r = 0; r < 8; ++r) {
        out[(long)(mbase + r) * D + n0 + row] = acc[r] + bias;
    }
}

// ----------------------------------------------------------------- launch
extern "C" void kernel_launch(void* const* d_in, const int* in_sizes, int n_in,
                              void* d_out, int out_size, void* d_ws, size_t ws_size,
                              hipStream_t stream) {
    const float* x     = (const float*)d_in[0];   // [N, 128] fp32
    const int*   eidx  = (const int*)d_in[1];     // [2, E] (harness: integer -> int*)
    const float* Wrel  = (const float*)d_in[2];   // [128, 128]
    const float* brel  = (const float*)d_in[3];   // [128]
    const float* Wroot = (const float*)d_in[4];   // [128, 128]
    float* out = (float*)d_out;                   // [N, 128] fp32
    float* agg = (float*)d_ws;                    // [N, 128] fp32 scratch

    const int N = in_sizes[0] / D;                // 100000 (divisible by 16)
    const int E = in_sizes[1] / 2;                // 600000

    const int n4 = N * D / 4;
    gnn_zero_kernel<<<(n4 + 255) / 256, 256, 0, stream>>>(agg, n4);

    const long tscatter = (long)E * 32;           // one wave per edge
    gnn_scatter_kernel<<<(int)((tscatter + 255) / 256), 256, 0, stream>>>(
        x, eidx, agg, E);

    gnn_gemm_kernel<<<N / 16, 256, 0, stream>>>(agg, x, Wrel, brel, Wroot, out);
}